// RNN_54107997995651
// MI455X (gfx1250) — compile-verified
//
#include <hip/hip_runtime.h>
#include <math.h>

// ---------------------------------------------------------------------------
// Leaky-integrator RNN on MI455X (gfx1250).
//   x_{t+1} = x + (1/tau)*(g*W@tanh(x) - x),  R[:,t] = tanh(x_t)
// N=3072, T=4096 sequential steps. W (37.7MB) lives in L2 (192MB); the run is
// L2-BW + sync-latency bound, so we use a persistent grid with one global
// barrier per step and fp32 WMMA (V_WMMA_F32_16X16X4_F32) for the mat-vec.
// ---------------------------------------------------------------------------

typedef __attribute__((ext_vector_type(2))) float v2f;
typedef __attribute__((ext_vector_type(8))) float v8f;

#define RNN_N        3072
#define WGS          96                 // persistent workgroups (must all be resident)
#define TPB          256                // 8 waves of 32 (wave32)
#define NWAVES       (TPB / 32)
#define ROWS_PER_WG  (RNN_N / WGS)      // 32 rows -> 2 row-blocks of 16
#define KSLICE       (RNN_N / NWAVES)   // 384 K-elements per wave
#define INV_TAU      0.2f               // 1/5.0
#define GAIN         1.5f

__global__ void rnn_init_ws(unsigned int* bar) {
    if (threadIdx.x == 0 && blockIdx.x == 0) *bar = 0u;  // d_ws is poisoned 0xAA
}

__launch_bounds__(TPB, 1)
__global__ void rnn_persistent(const float* __restrict__ W,
                               const float* __restrict__ x0,
                               float*       __restrict__ Rout,
                               float*       __restrict__ rbuf0,
                               float*       __restrict__ rbuf1,
                               unsigned int* __restrict__ bar,
                               int T, int ldR /* = T+1 */)
{
    __shared__ float r_lds[RNN_N];                   // 12 KB: current r vector
    __shared__ float partial[NWAVES][ROWS_PER_WG];   // 1 KB : per-wave partial dots
    __shared__ float x_lds[ROWS_PER_WG];             // this WG's slice of x state

    const int tid     = threadIdx.x;
    const int lane    = tid & 31;
    const int wave    = tid >> 5;
    const int rowbase = blockIdx.x * ROWS_PER_WG;

    // ---- step 0: load x0 slice, emit R[:,0] = tanh(x0), publish r into buf0 ----
    if (tid < ROWS_PER_WG) {
        float x = x0[rowbase + tid];
        x_lds[tid] = x;
        float r = tanhf(x);
        Rout[(size_t)(rowbase + tid) * ldR] = r;
        __hip_atomic_store(&rbuf0[rowbase + tid], r,
                           __ATOMIC_RELAXED, __HIP_MEMORY_SCOPE_AGENT);
    }
    __syncthreads();

    // ---- global barrier #0 (monotonic counter; all WGs resident) ----
    unsigned int barTarget = (unsigned int)gridDim.x;
    if (tid == 0) {
        __hip_atomic_fetch_add(bar, 1u, __ATOMIC_ACQ_REL, __HIP_MEMORY_SCOPE_AGENT);
        while (__hip_atomic_load(bar, __ATOMIC_ACQUIRE, __HIP_MEMORY_SCOPE_AGENT) < barTarget)
            __builtin_amdgcn_s_sleep(1);
    }
    __syncthreads();

    // A-matrix addressing per the documented 16x4 f32 layout:
    //   lane L: M = L%16, VGPR0 holds K = k0 + off, VGPR1 holds K = k0 + off + 1,
    //   with off = (L < 16) ? 0 : 2.  Same index pattern applies to B (r broadcast).
    const float* Wr0 = W + (size_t)(rowbase + (lane & 15)) * RNN_N;   // row-block 0
    const float* Wr1 = Wr0 + (size_t)16 * RNN_N;                      // row-block 1
    const int    off = (lane >> 4) << 1;                              // 0 or 2

    for (int t = 0; t < T; ++t) {
        const float* rsrc = (t & 1) ? rbuf1 : rbuf0;   // read  r_t
        float*       rdst = (t & 1) ? rbuf0 : rbuf1;   // write r_{t+1}

        // stage the full r vector into LDS (agent-scope loads: skip stale L0)
        for (int i = tid; i < RNN_N; i += TPB)
            r_lds[i] = __hip_atomic_load(&rsrc[i], __ATOMIC_RELAXED,
                                         __HIP_MEMORY_SCOPE_AGENT);
        __syncthreads();

        // ---- mat-vec: this wave covers K in [kbase, kbase+KSLICE) for 2 blocks
        v8f c0 = {}; v8f c1 = {};
        const int kbase = wave * KSLICE;
        #pragma unroll 4
        for (int k = kbase; k < kbase + KSLICE; k += 4) {
            v2f b  = *(const v2f*)&r_lds[k + off];       // B = r broadcast columns
            v2f a0 = *(const v2f*)(Wr0 + k + off);
            v2f a1 = *(const v2f*)(Wr1 + k + off);
            c0 = __builtin_amdgcn_wmma_f32_16x16x4_f32(false, a0, false, b,
                                                       (short)0, c0, false, false);
            c1 = __builtin_amdgcn_wmma_f32_16x16x4_f32(false, a1, false, b,
                                                       (short)0, c1, false, false);
        }

        // C layout: VGPR i, lanes 0-15 -> M=i (take lane 0); lanes 16-31 -> M=i+8
        if ((lane & 15) == 0) {
            const int mbase = (lane >> 4) * 8;           // 0 or 8
            #pragma unroll
            for (int i = 0; i < 8; ++i) {
                partial[wave][mbase + i]      = c0[i];
                partial[wave][16 + mbase + i] = c1[i];
            }
        }
        __syncthreads();

        // ---- reduce across waves, leaky update, emit R column, publish r ----
        if (tid < ROWS_PER_WG) {
            float acc = 0.f;
            #pragma unroll
            for (int w = 0; w < NWAVES; ++w) acc += partial[w][tid];
            float x = x_lds[tid];
            x = x + INV_TAU * (GAIN * acc - x);
            x_lds[tid] = x;
            float r = tanhf(x);
            Rout[(size_t)(rowbase + tid) * ldR + (t + 1)] = r;
            __hip_atomic_store(&rdst[rowbase + tid], r,
                               __ATOMIC_RELAXED, __HIP_MEMORY_SCOPE_AGENT);
        }
        __syncthreads();

        // ---- global barrier #(t+1): r_{t+1} fully published ----
        barTarget += (unsigned int)gridDim.x;
        if (tid == 0) {
            __hip_atomic_fetch_add(bar, 1u, __ATOMIC_ACQ_REL, __HIP_MEMORY_SCOPE_AGENT);
            while (__hip_atomic_load(bar, __ATOMIC_ACQUIRE, __HIP_MEMORY_SCOPE_AGENT) < barTarget)
                __builtin_amdgcn_s_sleep(1);
        }
        __syncthreads();
    }
}

extern "C" void kernel_launch(void* const* d_in, const int* in_sizes, int n_in,
                              void* d_out, int out_size, void* d_ws, size_t ws_size,
                              hipStream_t stream) {
    const float* W  = (const float*)d_in[0];   // (N, N) fp32 row-major
    const float* x0 = (const float*)d_in[1];   // (N,)  fp32
    // d_in[2] is T on *device*; derive it from out_size = N*(T+1) instead.
    const int ldR = out_size / RNN_N;          // T + 1
    const int T   = ldR - 1;

    float*        rbuf0 = (float*)d_ws;                  // double-buffered r vector
    float*        rbuf1 = rbuf0 + RNN_N;
    unsigned int* bar   = (unsigned int*)(rbuf1 + RNN_N);

    rnn_init_ws<<<1, 1, 0, stream>>>(bar);
    rnn_persistent<<<WGS, TPB, 0, stream>>>(W, x0, (float*)d_out,
                                            rbuf0, rbuf1, bar, T, ldR);
}